// GCN_65876208386063
// MI455X (gfx1250) — compile-verified
//
#include <hip/hip_runtime.h>
#include <math.h>

typedef __attribute__((ext_vector_type(2))) float v2f;
typedef __attribute__((ext_vector_type(8))) float v8f;

#define F_IN 512
#define HID  16
#define NCLS 7

// ---------------- degree / normalization ----------------

__global__ void k_init_deg(float* deg, int n) {
  int i = blockIdx.x * blockDim.x + threadIdx.x;
  if (i < n) deg[i] = 1.0f;  // self loop contributes 1
}

__global__ void k_degree(const int* __restrict__ row, float* deg, int e) {
  int i = blockIdx.x * blockDim.x + threadIdx.x;
  if (i < e) atomicAdd(&deg[row[i]], 1.0f);
}

__global__ void k_dis(float* deg, int n) {
  int i = blockIdx.x * blockDim.x + threadIdx.x;
  if (i < n) deg[i] = rsqrtf(deg[i]);  // deg >= 1 always
}

// ---------------- GEMM1: h = x @ W1  (N x 512 @ 512 x 16) ----------------
// One wave computes a 16-row tile via V_WMMA_F32_16X16X4_F32, K swept 0..512 by 4.
// x is a read-once 205MB stream (> 192MB L2) -> non-temporal loads so the
// edge-aggregation working set (h, edge indices) stays resident in L2.

__global__ void __launch_bounds__(256) k_gemm1(const float* __restrict__ x,
                                               const float* __restrict__ W1,
                                               float* __restrict__ h, int n) {
  __shared__ float w[F_IN * HID];  // 32 KB
  for (int i = threadIdx.x; i < F_IN * HID; i += blockDim.x) w[i] = W1[i];
  __syncthreads();

  const int wave = threadIdx.x >> 5;
  const int lane = threadIdx.x & 31;
  const int tile = blockIdx.x * (blockDim.x >> 5) + wave;
  const int row0 = tile * 16;
  if (row0 >= n) return;  // whole-wave guard: EXEC stays all-ones for WMMA

  const int m  = lane & 15;  // A: row within tile; B/D: column N
  const int hh = lane >> 4;  // K-half select {0,1} -> K offset {0,2}

  // 8-byte aligned: (row0+m)*512 + 2*hh is even
  const v2f* xp = (const v2f*)(x + (size_t)(row0 + m) * F_IN + 2 * hh);

  v8f acc0 = {};
  v8f acc1 = {};
  #pragma unroll 4
  for (int k = 0; k < F_IN; k += 8) {
    v2f a0 = __builtin_nontemporal_load(&xp[k >> 1]);
    v2f b0;
    b0.x = w[(k + 2 * hh) * HID + m];
    b0.y = w[(k + 2 * hh + 1) * HID + m];
    acc0 = __builtin_amdgcn_wmma_f32_16x16x4_f32(false, a0, false, b0,
                                                 (short)0, acc0, false, false);
    v2f a1 = __builtin_nontemporal_load(&xp[(k >> 1) + 2]);
    v2f b1;
    b1.x = w[(k + 4 + 2 * hh) * HID + m];
    b1.y = w[(k + 4 + 2 * hh + 1) * HID + m];
    acc1 = __builtin_amdgcn_wmma_f32_16x16x4_f32(false, a1, false, b1,
                                                 (short)0, acc1, false, false);
  }
  v8f acc = acc0 + acc1;

  // D element r lives at row (r + 8*hh), col m  -> contiguous 64B row stores
  float* out = h + (size_t)row0 * HID + m;
  #pragma unroll
  for (int r = 0; r < 8; ++r) out[(r + 8 * hh) * HID] = acc[r];
}

// ---------------- GEMM2: g = h1 @ W2pad (N x 16 @ 16 x 16, cols>=7 zero) ----------------

__global__ void __launch_bounds__(256) k_gemm2(const float* __restrict__ h1,
                                               const float* __restrict__ W2,
                                               float* __restrict__ g, int n) {
  __shared__ float w[HID * 16];
  {
    int t = threadIdx.x;
    if (t < HID * 16) {
      int k = t >> 4, c = t & 15;
      w[t] = (c < NCLS) ? W2[k * NCLS + c] : 0.0f;
    }
  }
  __syncthreads();

  const int wave = threadIdx.x >> 5;
  const int lane = threadIdx.x & 31;
  const int tile = blockIdx.x * (blockDim.x >> 5) + wave;
  const int row0 = tile * 16;
  if (row0 >= n) return;

  const int m  = lane & 15;
  const int hh = lane >> 4;

  const float* ar = h1 + (size_t)(row0 + m) * HID + 2 * hh;

  v8f acc = {};
  #pragma unroll
  for (int k = 0; k < HID; k += 4) {
    v2f a, b;
    a.x = ar[k];
    a.y = ar[k + 1];
    b.x = w[(k + 2 * hh) * 16 + m];
    b.y = w[(k + 2 * hh + 1) * 16 + m];
    acc = __builtin_amdgcn_wmma_f32_16x16x4_f32(false, a, false, b,
                                                (short)0, acc, false, false);
  }

  // keep only cols 0..7 (stride-8 output); col 7 is exactly 0
  if (m < 8) {
    float* out = g + (size_t)row0 * 8 + m;
    #pragma unroll
    for (int r = 0; r < 8; ++r) out[(r + 8 * hh) * 8] = acc[r];
  }
}

// ---------------- edge aggregation: out[col] += dis[row]*dis[col] * feat[row] ----------------

template <int F>
__global__ void k_agg(const int* __restrict__ rw, const int* __restrict__ cl,
                      const float* __restrict__ dis, const float* __restrict__ feat,
                      float* out, int e) {
  constexpr int SH = (F == 16) ? 4 : 3;
  int t = blockIdx.x * blockDim.x + threadIdx.x;
  int ed = t >> SH;
  int f  = t & (F - 1);
  if (ed >= e) return;
  int r = rw[ed], c = cl[ed];
  float nrm = dis[r] * dis[c];
  atomicAdd(&out[(size_t)c * F + f], nrm * feat[(size_t)r * F + f]);
}

// ---------------- bias + relu + self-loop fold (layer 1) ----------------

__global__ void k_bias_relu(float* __restrict__ agg, const float* __restrict__ h,
                            const float* __restrict__ dis, const float* __restrict__ b1,
                            int n) {
  int t = blockIdx.x * blockDim.x + threadIdx.x;
  int i = t >> 4, f = t & 15;
  if (i >= n) return;
  float d = dis[i];
  float v = agg[t] + d * d * h[t] + b1[f];
  agg[t] = v > 0.0f ? v : 0.0f;  // agg now holds h1
}

// ---------------- layer-2 bias + self loop + log_softmax ----------------

__global__ void k_final(const float* __restrict__ agg2, const float* __restrict__ g,
                        const float* __restrict__ dis, const float* __restrict__ b2,
                        float* __restrict__ out, int n) {
  int i = blockIdx.x * blockDim.x + threadIdx.x;
  if (i >= n) return;
  float d2 = dis[i] * dis[i];
  float z[NCLS];
  float mx = -1e30f;
  #pragma unroll
  for (int f = 0; f < NCLS; ++f) {
    z[f] = agg2[(size_t)i * 8 + f] + d2 * g[(size_t)i * 8 + f] + b2[f];
    mx = fmaxf(mx, z[f]);
  }
  float s = 0.0f;
  #pragma unroll
  for (int f = 0; f < NCLS; ++f) s += expf(z[f] - mx);
  float l = logf(s) + mx;
  #pragma unroll
  for (int f = 0; f < NCLS; ++f) out[(size_t)i * NCLS + f] = z[f] - l;
}

// ---------------- launch ----------------

extern "C" void kernel_launch(void* const* d_in, const int* in_sizes, int n_in,
                              void* d_out, int out_size, void* d_ws, size_t ws_size,
                              hipStream_t stream) {
  const float* x  = (const float*)d_in[0];
  const float* W1 = (const float*)d_in[1];
  const float* b1 = (const float*)d_in[2];
  const float* W2 = (const float*)d_in[3];
  const float* b2 = (const float*)d_in[4];
  const int*   ei = (const int*)d_in[5];

  const int n = in_sizes[0] / F_IN;
  const int e = in_sizes[5] / 2;
  const int* rw = ei;       // edge_index[0]
  const int* cl = ei + e;   // edge_index[1]

  char* p = (char*)d_ws;
  auto alloc = [&](size_t bytes) {
    char* r = p;
    p += (bytes + 255) & ~(size_t)255;
    return r;
  };
  float* dis  = (float*)alloc((size_t)n * 4);            // deg -> deg^-1/2
  float* h    = (float*)alloc((size_t)n * HID * 4);      // x @ W1
  float* agg1 = (float*)alloc((size_t)n * HID * 4);      // scatter, then h1
  float* g    = (float*)alloc((size_t)n * 8 * 4);        // h1 @ W2 (padded to 8)
  float* agg2 = (float*)alloc((size_t)n * 8 * 4);

  hipMemsetAsync(agg1, 0, (size_t)n * HID * 4, stream);
  hipMemsetAsync(agg2, 0, (size_t)n * 8 * 4, stream);

  const int B = 256;
  k_init_deg<<<(n + B - 1) / B, B, 0, stream>>>(dis, n);
  k_degree<<<(e + B - 1) / B, B, 0, stream>>>(rw, dis, e);
  k_dis<<<(n + B - 1) / B, B, 0, stream>>>(dis, n);

  const int tiles   = (n + 15) / 16;
  const int gblocks = (tiles + 7) / 8;  // 8 waves (tiles) per 256-thread block
  k_gemm1<<<gblocks, 256, 0, stream>>>(x, W1, h, n);

  {
    long long t = (long long)e * 16;
    k_agg<16><<<(int)((t + B - 1) / B), B, 0, stream>>>(rw, cl, dis, h, agg1, e);
  }
  {
    long long t = (long long)n * 16;
    k_bias_relu<<<(int)((t + B - 1) / B), B, 0, stream>>>(agg1, h, dis, b1, n);
  }

  k_gemm2<<<gblocks, 256, 0, stream>>>(agg1, W2, g, n);

  {
    long long t = (long long)e * 8;
    k_agg<8><<<(int)((t + B - 1) / B), B, 0, stream>>>(rw, cl, dis, g, agg2, e);
  }

  k_final<<<(n + B - 1) / B, B, 0, stream>>>(agg2, g, dis, b2, (float*)d_out, n);
}